// GAT_RW_full_13975823581632
// MI455X (gfx1250) — compile-verified
//
#include <hip/hip_runtime.h>
#include <hip/hip_bf16.h>
#include <math.h>

typedef float v2f __attribute__((ext_vector_type(2)));
typedef float v4f __attribute__((ext_vector_type(4)));
typedef float v8f __attribute__((ext_vector_type(8)));

#define GAT_N    50000
#define GAT_FIN  256
#define GAT_H    128
#define GAT_C    64
#define GAT_K    4
#define GAT_RWS  16

// ---------------------------------------------------------------------------
// GEMM: Y[16 rows x HOUT] = X_tile @ W^T + bias, via V_WMMA_F32_16X16X4_F32.
// Block = (HOUT/16) waves; each wave owns one 16-col tile, A staged in LDS.
// A layout per lane: m = lane&15, koff = (lane>>4)*2 -> {A[m][k+koff], A[m][k+koff+1]}
// B layout per lane: n = lane&15               -> {B[k+koff][n], B[k+koff+1][n]}
//                    and B[k][n] = W[col0+n][k]  (row of W, contiguous -> b64 load)
// C/D layout: VGPR r -> row (r + (lane>>4)*8), col = lane&15.
// ---------------------------------------------------------------------------
template <int DIN, int HOUT>
__global__ void gemm_bias_wmma(const float* __restrict__ X,
                               const float* __restrict__ W,
                               const float* __restrict__ B,
                               float* __restrict__ Y) {
  __shared__ float As[16 * DIN];
  const int tid = threadIdx.x;
  constexpr int NTHREADS = (HOUT / 16) * 32;
  const int row0 = blockIdx.x * 16;

  // Cooperatively stage the 16 x DIN A-tile into LDS (shared by all waves).
  const float* xrow = X + (size_t)row0 * DIN;
  #pragma unroll
  for (int i = tid; i < 16 * DIN; i += NTHREADS) As[i] = xrow[i];
  __syncthreads();

  const int wave = tid >> 5;
  const int lane = tid & 31;
  const int m    = lane & 15;
  const int koff = (lane >> 4) << 1;
  const int col0 = wave * 16;

  const float bv = B[col0 + m];
  v8f acc = {bv, bv, bv, bv, bv, bv, bv, bv};

  const float* wrow = W + (size_t)(col0 + m) * DIN + koff;
  const float* arow = As + m * DIN + koff;

  #pragma unroll 8
  for (int k = 0; k < DIN; k += 4) {
    v2f a = *(const v2f*)(arow + k);   // ds_load_b64
    v2f b = *(const v2f*)(wrow + k);   // global_load_b64 (L2-resident weights)
    acc = __builtin_amdgcn_wmma_f32_16x16x4_f32(
        /*neg_a=*/false, a, /*neg_b=*/false, b,
        /*c_mod=*/(short)0, acc, /*reuse_a=*/false, /*reuse_b=*/false);
  }

  const int mrow = (lane >> 4) << 3;  // 0 or 8
  float* yout = Y + (size_t)(row0 + mrow) * HOUT + col0 + m;
  #pragma unroll
  for (int r = 0; r < 8; ++r) yout[(size_t)r * HOUT] = acc[r];
}

// ---------------------------------------------------------------------------
// Attention scalars: al[i] = h[i].attl_w + attl_b ; ar[i] = h[i].attr_w + attr_b
// One wave per node (H=128 -> float4 per lane), width-32 shuffle reduction.
// ---------------------------------------------------------------------------
__global__ void att_kernel(const float* __restrict__ h,
                           const float* __restrict__ alw, const float* __restrict__ albp,
                           const float* __restrict__ arw, const float* __restrict__ arbp,
                           float* __restrict__ al, float* __restrict__ ar) {
  const int node = blockIdx.x * 8 + (threadIdx.x >> 5);
  const int lane = threadIdx.x & 31;

  v4f hv = *(const v4f*)(h + (size_t)node * GAT_H + lane * 4);
  v4f wl = *(const v4f*)(alw + lane * 4);
  v4f wr = *(const v4f*)(arw + lane * 4);
  float sl = hv.x * wl.x + hv.y * wl.y + hv.z * wl.z + hv.w * wl.w;
  float sr = hv.x * wr.x + hv.y * wr.y + hv.z * wr.z + hv.w * wr.w;

  #pragma unroll
  for (int msk = 16; msk; msk >>= 1) {
    sl += __shfl_xor(sl, msk, 32);
    sr += __shfl_xor(sr, msk, 32);
  }
  if (lane == 0) {
    al[node] = sl + *albp;
    ar[node] = sr + *arbp;
  }
}

// ---------------------------------------------------------------------------
// Sampled attention aggregation, one wave per node.
//   agg = att[0]*h[i] + sum_k att[k+1] * mean_j( softmax_j(lrelu(al_i+ar_ej)) * h[e_j] )
// Lanes 0..15 handle the 16 walk samples (logits + width-16 shuffle softmax);
// all 32 lanes accumulate the gathered h rows (float4/lane, L2-resident).
// ---------------------------------------------------------------------------
__global__ void aggr_kernel(const float* __restrict__ h,
                            const float* __restrict__ al, const float* __restrict__ ar,
                            const int* __restrict__ ends_l,   // [K][N*RWS]
                            const float* __restrict__ att_l,  // [K+1]
                            float* __restrict__ agg, int n) {
  const int node = blockIdx.x * 8 + (threadIdx.x >> 5);
  const int lane = threadIdx.x & 31;

  v4f hi = *(const v4f*)(h + (size_t)node * GAT_H + lane * 4);
  v4f acc = hi * att_l[0];
  const float ali = al[node];

  #pragma unroll
  for (int k = 0; k < GAT_K; ++k) {
    const int* e = ends_l + (size_t)k * ((size_t)n * GAT_RWS) + (size_t)node * GAT_RWS;

    int   ej    = 0;
    float logit = -1e30f;
    if (lane < 16) {
      ej = e[lane];
      float z = ali + ar[ej];
      logit = (z > 0.0f) ? z : 0.2f * z;  // leaky_relu(0.2)
    }
    // softmax over the 16 samples (confined to lanes 0..15 via width-16 shfl)
    float mx = logit;
    mx = fmaxf(mx, __shfl_xor(mx, 8, 16));
    mx = fmaxf(mx, __shfl_xor(mx, 4, 16));
    mx = fmaxf(mx, __shfl_xor(mx, 2, 16));
    mx = fmaxf(mx, __shfl_xor(mx, 1, 16));
    float we = (lane < 16) ? expf(logit - mx) : 0.0f;
    float s = we;
    s += __shfl_xor(s, 8, 16);
    s += __shfl_xor(s, 4, 16);
    s += __shfl_xor(s, 2, 16);
    s += __shfl_xor(s, 1, 16);
    float wn = we / s;

    v4f tmp = {0.0f, 0.0f, 0.0f, 0.0f};
    #pragma unroll
    for (int j = 0; j < GAT_RWS; ++j) {
      float wj  = __shfl(wn, j, 32);
      int   ejb = __shfl(ej, j, 32);
      v4f hv = *(const v4f*)(h + (size_t)ejb * GAT_H + lane * 4);
      tmp += hv * wj;
    }
    acc += tmp * (att_l[k + 1] * (1.0f / GAT_RWS));
  }

  *(v4f*)(agg + (size_t)node * GAT_H + lane * 4) = acc;
}

// ---------------------------------------------------------------------------
// Output head: out = log_softmax(X @ lout_w^T + lout_b), fused.
// 4 waves (one per 16-col tile of C=64) do WMMA into LDS, then 8 threads/row
// compute the row log-sum-exp with width-8 shuffle reductions.
// ---------------------------------------------------------------------------
__global__ void out_lsm_kernel(const float* __restrict__ X,   // [N,128]
                               const float* __restrict__ W,   // [64,128]
                               const float* __restrict__ B,   // [64]
                               float* __restrict__ out) {     // [N,64]
  __shared__ float As[16 * GAT_H];
  __shared__ float Os[16][GAT_C];
  const int tid = threadIdx.x;
  const int row0 = blockIdx.x * 16;

  const float* xrow = X + (size_t)row0 * GAT_H;
  #pragma unroll
  for (int i = tid; i < 16 * GAT_H; i += 128) As[i] = xrow[i];
  __syncthreads();

  const int wave = tid >> 5;
  const int lane = tid & 31;
  const int m    = lane & 15;
  const int koff = (lane >> 4) << 1;
  const int col0 = wave * 16;

  const float bv = B[col0 + m];
  v8f acc = {bv, bv, bv, bv, bv, bv, bv, bv};
  const float* wrow = W + (size_t)(col0 + m) * GAT_H + koff;
  const float* arow = As + m * GAT_H + koff;

  #pragma unroll 8
  for (int k = 0; k < GAT_H; k += 4) {
    v2f a = *(const v2f*)(arow + k);
    v2f b = *(const v2f*)(wrow + k);
    acc = __builtin_amdgcn_wmma_f32_16x16x4_f32(false, a, false, b, (short)0, acc,
                                                false, false);
  }

  const int mrow = (lane >> 4) << 3;
  #pragma unroll
  for (int r = 0; r < 8; ++r) Os[mrow + r][col0 + m] = acc[r];
  __syncthreads();

  // log_softmax: 8 threads per row, 8 contiguous cols each.
  const int row = tid >> 3;
  const int sub = tid & 7;
  float mx = -1e30f;
  #pragma unroll
  for (int c = 0; c < 8; ++c) mx = fmaxf(mx, Os[row][sub * 8 + c]);
  mx = fmaxf(mx, __shfl_xor(mx, 4, 8));
  mx = fmaxf(mx, __shfl_xor(mx, 2, 8));
  mx = fmaxf(mx, __shfl_xor(mx, 1, 8));
  float se = 0.0f;
  #pragma unroll
  for (int c = 0; c < 8; ++c) se += expf(Os[row][sub * 8 + c] - mx);
  se += __shfl_xor(se, 4, 8);
  se += __shfl_xor(se, 2, 8);
  se += __shfl_xor(se, 1, 8);
  const float lse = mx + logf(se);

  float* orow = out + (size_t)(row0 + row) * GAT_C + sub * 8;
  #pragma unroll
  for (int c = 0; c < 8; ++c) orow[c] = Os[row][sub * 8 + c] - lse;
}

// ---------------------------------------------------------------------------
extern "C" void kernel_launch(void* const* d_in, const int* in_sizes, int n_in,
                              void* d_out, int out_size, void* d_ws, size_t ws_size,
                              hipStream_t stream) {
  const float* x      = (const float*)d_in[0];
  const int*   ends   = (const int*)d_in[1];     // [NLAYER, K, N*RWS]
  const float* lin0_w = (const float*)d_in[2];
  const float* lin0_b = (const float*)d_in[3];
  const float* lin1_w = (const float*)d_in[4];
  const float* lin1_b = (const float*)d_in[5];
  const float* lout_w = (const float*)d_in[6];
  const float* lout_b = (const float*)d_in[7];
  const float* attl_w = (const float*)d_in[8];   // [NLAYER, H]
  const float* attl_b = (const float*)d_in[9];   // [NLAYER]
  const float* attr_w = (const float*)d_in[10];
  const float* attr_b = (const float*)d_in[11];
  const float* att    = (const float*)d_in[12];  // [NLAYER, K+1]
  float* out = (float*)d_out;

  constexpr int N = GAT_N, H = GAT_H, K = GAT_K, RWS = GAT_RWS;

  // Workspace layout: h [N,H], agg [N,H], al [N], ar [N]  (~52 MB)
  float* h   = (float*)d_ws;
  float* agg = h + (size_t)N * H;
  float* al  = agg + (size_t)N * H;
  float* ar  = al + N;

  const int gemmBlocks = N / 16;  // 3125
  const int nodeBlocks = N / 8;   // 6250

  // -------- Layer 0 --------
  gemm_bias_wmma<GAT_FIN, GAT_H><<<gemmBlocks, 256, 0, stream>>>(x, lin0_w, lin0_b, h);
  att_kernel<<<nodeBlocks, 256, 0, stream>>>(h, attl_w, attl_b, attr_w, attr_b, al, ar);
  aggr_kernel<<<nodeBlocks, 256, 0, stream>>>(h, al, ar, ends, att, agg, N);

  // -------- Layer 1 --------
  gemm_bias_wmma<GAT_H, GAT_H><<<gemmBlocks, 256, 0, stream>>>(agg, lin1_w, lin1_b, h);
  att_kernel<<<nodeBlocks, 256, 0, stream>>>(h, attl_w + H, attl_b + 1,
                                             attr_w + H, attr_b + 1, al, ar);
  aggr_kernel<<<nodeBlocks, 256, 0, stream>>>(h, al, ar,
                                              ends + (size_t)K * N * RWS,
                                              att + (K + 1), agg, N);

  // -------- Output head + log_softmax --------
  out_lsm_kernel<<<gemmBlocks, 128, 0, stream>>>(agg, lout_w, lout_b, out);
}